// GNNClassifier_26834955665908
// MI455X (gfx1250) — compile-verified
//
#include <hip/hip_runtime.h>
#include <math.h>

#define N_NODES 100000
#define E_EDGES 3200000
#define G_GRAPHS 5000
#define NHEADS 8
#define HC 16
#define BN_EPS 1e-5f
#define ET (E_EDGES + N_NODES)  // edges incl. self-loops

typedef __attribute__((ext_vector_type(2))) float v2f;
typedef __attribute__((ext_vector_type(8))) float v8f;

// ---------------- utility ----------------
__global__ void fill_kernel(float* __restrict__ p, float v, int n) {
  int i = blockIdx.x * blockDim.x + threadIdx.x;
  if (i < n) p[i] = v;
}

__device__ __forceinline__ void atomicMaxF(float* addr, float val) {
  int* ai = (int*)addr;
  int old = __float_as_int(*addr);
  while (__int_as_float(old) < val) {
    int assumed = old;
    old = atomicCAS(ai, assumed, __float_as_int(val));
    if (old == assumed) break;
  }
}

__device__ __forceinline__ float leaky02(float x) { return x > 0.f ? x : 0.2f * x; }

// ---------------- degree / norm ----------------
__global__ void deg_kernel(const int* __restrict__ dst, float* __restrict__ deg) {
  int e = blockIdx.x * blockDim.x + threadIdx.x;
  if (e >= ET) return;
  int d = (e < E_EDGES) ? dst[e] : (e - E_EDGES);  // self-loop tail
  atomicAdd(&deg[d], 1.0f);
}

__global__ void dinv_kernel(float* __restrict__ deg) {
  int i = blockIdx.x * blockDim.x + threadIdx.x;
  if (i >= N_NODES) return;
  float d = deg[i];
  deg[i] = (d > 0.f) ? rsqrtf(d) : 0.f;
}

// ---------------- WMMA f32 GEMM: C[MxN] = A[MxK] @ B[KxN], row-major ----------------
// M%16==0, N%16==0, K%4==0. One wave computes one 16x16 tile with V_WMMA_F32_16X16X4_F32.
__global__ void gemm_wmma_f32(const float* __restrict__ A, const float* __restrict__ B,
                              float* __restrict__ C, int M, int K, int N) {
  int wave = (blockIdx.x * blockDim.x + threadIdx.x) >> 5;
  int lane = threadIdx.x & 31;
  int tilesN = N >> 4;
  int tilesM = M >> 4;
  if (wave >= tilesM * tilesN) return;  // uniform per wave -> EXEC all-1s inside
  int tm = wave / tilesN;
  int tn = wave % tilesN;
  int half = lane >> 4;   // lane group 0/1
  int l16  = lane & 15;
  int m = tm * 16 + l16;  // A row for this lane
  int n = tn * 16 + l16;  // B/C column for this lane
  v8f c = {0.f, 0.f, 0.f, 0.f, 0.f, 0.f, 0.f, 0.f};
  for (int k0 = 0; k0 < K; k0 += 4) {
    int ka = k0 + 2 * half;
    v2f a, b;
    a.x = A[(long)m * K + ka];
    a.y = A[(long)m * K + ka + 1];
    b.x = B[(long)ka * N + n];
    b.y = B[(long)(ka + 1) * N + n];
    // (neg_a, A, neg_b, B, c_mod, C, reuse_a, reuse_b)
    c = __builtin_amdgcn_wmma_f32_16x16x4_f32(false, a, false, b, (short)0, c, false, false);
  }
#pragma unroll
  for (int j = 0; j < 8; ++j) {
    C[(long)(tm * 16 + j + 8 * half) * N + n] = c[j];
  }
}

// ---------------- GCN scatter: out[dst] += t[src] * dinv[src]*dinv[dst] ----------------
__global__ void gcn_scatter(const float* __restrict__ t, const int* __restrict__ src,
                            const int* __restrict__ dst, const float* __restrict__ dinv,
                            float* __restrict__ out, int F) {
  int e = blockIdx.x * blockDim.x + threadIdx.x;
  if (e >= ET) return;
  int s, d;
  if (e < E_EDGES) { s = src[e]; d = dst[e]; } else { s = d = e - E_EDGES; }
  float norm = dinv[s] * dinv[d];
  const float* row = t + (long)s * F;
  float* orow = out + (long)d * F;
  for (int f = 0; f < F; ++f) atomicAdd(&orow[f], row[f] * norm);
}

// ---------------- BatchNorm (train-mode stats) + ReLU ----------------
// stats layout: [0,F) sum, [F,2F) sumsq, [2F,3F) scale, [3F,4F) shift
__global__ void bn_stats(const float* __restrict__ h, float* __restrict__ stats,
                         int F, int rowsPerBlock) {
  __shared__ float ssum[128], ssq[128];
  int tid = threadIdx.x;
  if (tid < F) { ssum[tid] = 0.f; ssq[tid] = 0.f; }
  __syncthreads();
  int f = tid % F;
  int rlane = tid / F;
  int rstep = blockDim.x / F;
  int r0 = blockIdx.x * rowsPerBlock;
  int r1 = r0 + rowsPerBlock; if (r1 > N_NODES) r1 = N_NODES;
  float s = 0.f, q = 0.f;
  for (int r = r0 + rlane; r < r1; r += rstep) {
    float v = h[(long)r * F + f];
    s += v; q += v * v;
  }
  atomicAdd(&ssum[f], s);
  atomicAdd(&ssq[f], q);
  __syncthreads();
  if (tid < F) {
    atomicAdd(&stats[f], ssum[tid]);
    atomicAdd(&stats[F + f], ssq[tid]);
  }
}

// GCN bias b cancels exactly in train-mode BN (mean subtraction), so it is omitted.
__global__ void bn_finalize(float* __restrict__ stats, const float* __restrict__ g,
                            const float* __restrict__ be, int F) {
  int f = threadIdx.x;
  if (f >= F) return;
  float inv_n = 1.0f / (float)N_NODES;
  float mu  = stats[f] * inv_n;
  float var = stats[F + f] * inv_n - mu * mu;
  float sc  = g[f] * rsqrtf(var + BN_EPS);
  stats[2 * F + f] = sc;
  stats[3 * F + f] = be[f] - mu * sc;
}

__global__ void bn_relu_apply(float* __restrict__ h, const float* __restrict__ stats, int F) {
  int i = blockIdx.x * blockDim.x + threadIdx.x;
  if (i >= N_NODES * F) return;
  int f = i % F;
  float v = h[i] * stats[2 * F + f] + stats[3 * F + f];
  h[i] = v > 0.f ? v : 0.f;
}

// ---------------- GAT ----------------
__global__ void att_scores(const float* __restrict__ xg, const float* __restrict__ att_src,
                           const float* __restrict__ att_dst, float* __restrict__ asrc,
                           float* __restrict__ adst) {
  int i = blockIdx.x * blockDim.x + threadIdx.x;  // node*NHEADS + h
  if (i >= N_NODES * NHEADS) return;
  int n = i / NHEADS, h = i % NHEADS;
  const float* row = xg + (long)n * NHEADS * HC + h * HC;
  float s1 = 0.f, s2 = 0.f;
  for (int c = 0; c < HC; ++c) {
    float v = row[c];
    s1 += v * att_src[h * HC + c];
    s2 += v * att_dst[h * HC + c];
  }
  asrc[i] = s1;
  adst[i] = s2;
}

__global__ void edge_max(const int* __restrict__ src, const int* __restrict__ dst,
                         const float* __restrict__ asrc, const float* __restrict__ adst,
                         float* __restrict__ m) {
  int e = blockIdx.x * blockDim.x + threadIdx.x;
  if (e >= ET) return;
  int s, d;
  if (e < E_EDGES) { s = src[e]; d = dst[e]; } else { s = d = e - E_EDGES; }
  for (int h = 0; h < NHEADS; ++h) {
    float logit = leaky02(asrc[s * NHEADS + h] + adst[d * NHEADS + h]);
    atomicMaxF(&m[d * NHEADS + h], logit);
  }
}

__global__ void edge_expsum(const int* __restrict__ src, const int* __restrict__ dst,
                            const float* __restrict__ asrc, const float* __restrict__ adst,
                            const float* __restrict__ m, float* __restrict__ ssum) {
  int e = blockIdx.x * blockDim.x + threadIdx.x;
  if (e >= ET) return;
  int s, d;
  if (e < E_EDGES) { s = src[e]; d = dst[e]; } else { s = d = e - E_EDGES; }
  for (int h = 0; h < NHEADS; ++h) {
    float logit = leaky02(asrc[s * NHEADS + h] + adst[d * NHEADS + h]);
    atomicAdd(&ssum[d * NHEADS + h], expf(logit - m[d * NHEADS + h]));
  }
}

__global__ void edge_agg(const int* __restrict__ src, const int* __restrict__ dst,
                         const float* __restrict__ asrc, const float* __restrict__ adst,
                         const float* __restrict__ m, const float* __restrict__ ssum,
                         const float* __restrict__ xg, float* __restrict__ agg) {
  int e = blockIdx.x * blockDim.x + threadIdx.x;
  if (e >= ET) return;
  int s, d;
  if (e < E_EDGES) { s = src[e]; d = dst[e]; } else { s = d = e - E_EDGES; }
  for (int h = 0; h < NHEADS; ++h) {
    float logit = leaky02(asrc[s * NHEADS + h] + adst[d * NHEADS + h]);
    float alpha = expf(logit - m[d * NHEADS + h]) / ssum[d * NHEADS + h];
    const float* xrow = xg + (long)s * (NHEADS * HC) + h * HC;
    float* arow = agg + (long)d * (NHEADS * HC) + h * HC;
    for (int c = 0; c < HC; ++c) atomicAdd(&arow[c], xrow[c] * alpha);
  }
}

// hg = mean over heads + bg, then global_add_pool into pooled[batch[n]]
__global__ void head_mean_pool(const float* __restrict__ agg, const float* __restrict__ bg,
                               const int* __restrict__ batch, float* __restrict__ pooled) {
  int i = blockIdx.x * blockDim.x + threadIdx.x;  // node*HC + c
  if (i >= N_NODES * HC) return;
  int n = i / HC, c = i % HC;
  float s = 0.f;
  for (int h = 0; h < NHEADS; ++h) s += agg[(long)n * (NHEADS * HC) + h * HC + c];
  float hg = s * (1.0f / NHEADS) + bg[c];
  atomicAdd(&pooled[batch[n] * HC + c], hg);
}

__global__ void final_linear(const float* __restrict__ pooled, const float* __restrict__ Wf,
                             const float* __restrict__ bf, float* __restrict__ out) {
  int i = blockIdx.x * blockDim.x + threadIdx.x;  // g*3 + o
  if (i >= G_GRAPHS * 3) return;
  int g = i / 3, o = i % 3;
  float s = bf[o];
  for (int c = 0; c < 16; ++c) s += pooled[g * 16 + c] * Wf[c * 3 + o];
  out[i] = s;
}

// ---------------- launcher ----------------
extern "C" void kernel_launch(void* const* d_in, const int* in_sizes, int n_in,
                              void* d_out, int out_size, void* d_ws, size_t ws_size,
                              hipStream_t stream) {
  const float* x    = (const float*)d_in[0];
  const int*   ei   = (const int*)d_in[1];
  const int*   batch= (const int*)d_in[2];
  const float* W1   = (const float*)d_in[3];
  const float* g1   = (const float*)d_in[5];
  const float* be1  = (const float*)d_in[6];
  const float* W2   = (const float*)d_in[7];
  const float* g2   = (const float*)d_in[9];
  const float* be2  = (const float*)d_in[10];
  const float* W3   = (const float*)d_in[11];
  const float* g3   = (const float*)d_in[13];
  const float* be3  = (const float*)d_in[14];
  const float* Wg   = (const float*)d_in[15];
  const float* attS = (const float*)d_in[16];
  const float* attD = (const float*)d_in[17];
  const float* bg   = (const float*)d_in[18];
  const float* Wf   = (const float*)d_in[19];
  const float* bf   = (const float*)d_in[20];
  float* out = (float*)d_out;

  const int* src = ei;             // edge_index[0]
  const int* dst = ei + E_EDGES;   // edge_index[1]

  // workspace layout (floats)
  float* ws   = (float*)d_ws;
  float* T    = ws;                          // N*128: t1/t2/t3, then GAT agg
  float* A1   = T  + (size_t)N_NODES * 128;  // N*64: h1, then asrc|adst|m|s (4 x N*8)
  float* A2   = A1 + (size_t)N_NODES * 64;   // N*32: h2
  float* A3   = A2 + (size_t)N_NODES * 32;   // N*16: h3
  float* X    = A3 + (size_t)N_NODES * 16;   // N*128: xg
  float* deg  = X  + (size_t)N_NODES * 128;  // N
  float* stats= deg + N_NODES;               // 512
  float* pooled = stats + 512;               // G*16
  float* asrc = A1;
  float* adst = A1 + (size_t)N_NODES * 8;
  float* mmax = A1 + (size_t)N_NODES * 16;
  float* ssum = A1 + (size_t)N_NODES * 24;

  auto fill = [&](float* p, float v, long n) {
    fill_kernel<<<(int)((n + 255) / 256), 256, 0, stream>>>(p, v, (int)n);
  };
  auto gemm = [&](const float* A, const float* B, float* C, int M, int K, int Nc) {
    int waves = (M / 16) * (Nc / 16);
    gemm_wmma_f32<<<(waves + 7) / 8, 256, 0, stream>>>(A, B, C, M, K, Nc);
  };
  const int EB = (ET + 255) / 256;

  // ---- degree / normalization ----
  fill(deg, 0.f, N_NODES);
  deg_kernel<<<EB, 256, 0, stream>>>(dst, deg);
  dinv_kernel<<<(N_NODES + 255) / 256, 256, 0, stream>>>(deg);

  // ---- GCN layer 1: 20 -> 64 ----
  gemm(x, W1, T, N_NODES, 20, 64);
  fill(A1, 0.f, (long)N_NODES * 64);
  gcn_scatter<<<EB, 256, 0, stream>>>(T, src, dst, deg, A1, 64);
  fill(stats, 0.f, 256);
  bn_stats<<<(N_NODES + 1023) / 1024, 256, 0, stream>>>(A1, stats, 64, 1024);
  bn_finalize<<<1, 128, 0, stream>>>(stats, g1, be1, 64);
  bn_relu_apply<<<(N_NODES * 64 + 255) / 256, 256, 0, stream>>>(A1, stats, 64);

  // ---- GCN layer 2: 64 -> 32 ----
  gemm(A1, W2, T, N_NODES, 64, 32);
  fill(A2, 0.f, (long)N_NODES * 32);
  gcn_scatter<<<EB, 256, 0, stream>>>(T, src, dst, deg, A2, 32);
  fill(stats, 0.f, 256);
  bn_stats<<<(N_NODES + 1023) / 1024, 256, 0, stream>>>(A2, stats, 32, 1024);
  bn_finalize<<<1, 128, 0, stream>>>(stats, g2, be2, 32);
  bn_relu_apply<<<(N_NODES * 32 + 255) / 256, 256, 0, stream>>>(A2, stats, 32);

  // ---- GCN layer 3: 32 -> 16 ----
  gemm(A2, W3, T, N_NODES, 32, 16);
  fill(A3, 0.f, (long)N_NODES * 16);
  gcn_scatter<<<EB, 256, 0, stream>>>(T, src, dst, deg, A3, 16);
  fill(stats, 0.f, 256);
  bn_stats<<<(N_NODES + 1023) / 1024, 256, 0, stream>>>(A3, stats, 16, 1024);
  bn_finalize<<<1, 128, 0, stream>>>(stats, g3, be3, 16);
  bn_relu_apply<<<(N_NODES * 16 + 255) / 256, 256, 0, stream>>>(A3, stats, 16);

  // ---- GAT: xg = h3 @ Wg  (16 -> 8 heads x 16) ----
  gemm(A3, Wg, X, N_NODES, 16, 128);
  att_scores<<<(N_NODES * NHEADS + 255) / 256, 256, 0, stream>>>(X, attS, attD, asrc, adst);
  fill(mmax, -INFINITY, (long)N_NODES * NHEADS);
  fill(ssum, 0.f, (long)N_NODES * NHEADS);
  fill(T, 0.f, (long)N_NODES * 128);  // agg accumulator (T free after gemm3 scatter)
  edge_max<<<EB, 256, 0, stream>>>(src, dst, asrc, adst, mmax);
  edge_expsum<<<EB, 256, 0, stream>>>(src, dst, asrc, adst, mmax, ssum);
  edge_agg<<<EB, 256, 0, stream>>>(src, dst, asrc, adst, mmax, ssum, X, T);

  // ---- mean over heads + bias + global_add_pool ----
  fill(pooled, 0.f, (long)G_GRAPHS * 16);
  head_mean_pool<<<(N_NODES * HC + 255) / 256, 256, 0, stream>>>(T, bg, batch, pooled);

  // ---- final classifier ----
  final_linear<<<(G_GRAPHS * 3 + 255) / 256, 256, 0, stream>>>(pooled, Wf, bf, out);
}